// GraphSageLayer_22497038697225
// MI455X (gfx1250) — compile-verified
//
#include <hip/hip_runtime.h>
#include <hip/hip_bf16.h>

// Problem constants (match the reference)
#define R_REL   5
#define U_N     100000
#define M_N     50000
#define E_N     2000000
#define F_DIM   10
#define D_DIM   50
#define OUT_STRIDE (R_REL * D_DIM)   // 250

typedef float v2f __attribute__((ext_vector_type(2)));
typedef float v8f __attribute__((ext_vector_type(8)));

// ---------------------------------------------------------------------------
// Phase 1: edge scatter. One thread per (edge, direction).
//   dir 0: gather feat_u_fwd[r][src], atomic-add into s_fwd[r][dst] (+count)
//   dir 1: gather feat_i_rev[r][dst], atomic-add into s_rev[r][src] (+count)
// Accumulators (33MB) + feature tables (30MB) stay L2-resident (192MB L2);
// edge indices (80MB) are streamed once -> non-temporal loads so they don't
// evict the RMW working set.
// ---------------------------------------------------------------------------
__global__ __launch_bounds__(256) void edge_scatter_kernel(
    const int* __restrict__ edges_src, const int* __restrict__ edges_dst,
    const float* __restrict__ feat_u_fwd, const float* __restrict__ feat_i_rev,
    float* __restrict__ ws)
{
    int e = blockIdx.x * blockDim.x + threadIdx.x;
    if (e >= E_N) return;
    int y   = blockIdx.y;
    int r   = y >> 1;
    int dir = y & 1;

    int u = __builtin_nontemporal_load(edges_src + (size_t)r * E_N + e);
    int i = __builtin_nontemporal_load(edges_dst + (size_t)r * E_N + e);

    float* s_fwd   = ws;
    float* cnt_fwd = s_fwd + (size_t)R_REL * M_N * F_DIM;
    float* s_rev   = cnt_fwd + (size_t)R_REL * M_N;
    float* cnt_rev = s_rev + (size_t)R_REL * U_N * F_DIM;

    if (dir == 0) {
        const float* f = feat_u_fwd + ((size_t)r * U_N + u) * F_DIM;
        float*       s = s_fwd      + ((size_t)r * M_N + i) * F_DIM;
        #pragma unroll
        for (int k = 0; k < F_DIM; k += 2) {
            float2 v = *reinterpret_cast<const float2*>(f + k);  // 8B-aligned (40B rows)
            atomicAdd(s + k,     v.x);
            atomicAdd(s + k + 1, v.y);
        }
        atomicAdd(cnt_fwd + (size_t)r * M_N + i, 1.0f);
    } else {
        const float* f = feat_i_rev + ((size_t)r * M_N + i) * F_DIM;
        float*       s = s_rev      + ((size_t)r * U_N + u) * F_DIM;
        #pragma unroll
        for (int k = 0; k < F_DIM; k += 2) {
            float2 v = *reinterpret_cast<const float2*>(f + k);
            atomicAdd(s + k,     v.x);
            atomicAdd(s + k + 1, v.y);
        }
        atomicAdd(cnt_rev + (size_t)r * U_N + u, 1.0f);
    }
}

// ---------------------------------------------------------------------------
// Phase 2: per-node dense transform as WMMA f32 16x16x4 GEMM.
//   out[row, r*50+n] = sum_k A[row,k] * W[k,n] + bias[n]
//   A[row, 0:10]  = feat_dst[r][row][:]          (fc_self input)
//   A[row,10:20]  = s[r][row][:] / max(cnt,1)    (fc_neigh input, mean)
//   W[0:10 ,n] = w_self[r][:,n], W[10:20,n] = w_neigh[r][:,n]
// Weights staged in LDS PAIR-INTERLEAVED: wlds[kp][n][v] = W[2*kp+v][n]
// (n padded to 64), so a lane's B operand pair is one contiguous 8B
// ds_load_b64 -> lands directly in an even-aligned VGPR pair for WMMA.
// K=20 -> 5 wmma k-steps; N=50 -> 4 tiles of 16.
// 128 threads = 4 waves; each wave owns a 16-row tile (64 rows / block).
// Out-of-range rows are CLAMPED (loads always valid, stores guarded) so all
// A loads are branchless float2 loads and EXEC is all-1s at every WMMA.
// ---------------------------------------------------------------------------
__global__ __launch_bounds__(128) void sage_wmma_kernel(
    const float* __restrict__ feat_dst, const float* __restrict__ s_sum,
    const float* __restrict__ cnt,
    const float* __restrict__ w_self, const float* __restrict__ w_neigh,
    const float* __restrict__ bias,
    float* __restrict__ out, int nnode)
{
    __shared__ float wlds[10 * 64 * 2];   // [kp][n][v], kp=k/2, v=k&1
    __shared__ float blds[64];

    const int r = blockIdx.y;
    const int t = threadIdx.x;

    // Stage pair-interleaved padded weights + bias into LDS.
    for (int idx = t; idx < 10 * 64 * 2; idx += 128) {
        int kp = idx >> 7;            // 0..9
        int n  = (idx >> 1) & 63;     // 0..63
        int v  = idx & 1;             // 0..1
        int k  = kp * 2 + v;          // 0..19
        float w = 0.0f;
        if (n < D_DIM) {
            w = (k < F_DIM)
                    ? w_self [((size_t)r * F_DIM + k)           * D_DIM + n]
                    : w_neigh[((size_t)r * F_DIM + (k - F_DIM)) * D_DIM + n];
        }
        wlds[idx] = w;
    }
    for (int idx = t; idx < 64; idx += 128)
        blds[idx] = (idx < D_DIM) ? bias[(size_t)r * D_DIM + idx] : 0.0f;
    __syncthreads();

    const int wave = t >> 5;
    const int lane = t & 31;
    const int half = lane >> 4;     // 0: lanes 0-15, 1: lanes 16-31
    const int rIdx = lane & 15;
    const int rowBase = blockIdx.x * 64 + wave * 16;
    const int row  = rowBase + rIdx;
    const int rowC = (row < nnode) ? row : (nnode - 1);   // clamp: loads always valid

    const size_t nbase = (size_t)r * nnode + rowC;
    const float* fd = feat_dst + nbase * F_DIM;
    const float* sp = s_sum    + nbase * F_DIM;
    const float  cN = cnt[nbase];
    const float  inv = 1.0f / fmaxf(cN, 1.0f);

    const v2f* wp = reinterpret_cast<const v2f*>(wlds);   // v2f units: [kp][n]

    v8f acc[4] = {};   // bias added in epilogue

    #pragma unroll
    for (int ki = 0; ki < 5; ++ki) {
        // A operand: lane holds A[row, kb + {0,1}], kb = ki*4 + 2*half (even).
        // The pair never straddles the k=10 boundary -> single float2 source.
        const int kb = ki * 4 + half * 2;
        const int kp = ki * 2 + half;          // kb / 2
        const float* asrc = (kb < F_DIM) ? (fd + kb) : (sp + (kb - F_DIM));
        const float  sc   = (kb < F_DIM) ? 1.0f : inv;
        float2 av = *reinterpret_cast<const float2*>(asrc);
        v2f a; a[0] = av.x * sc; a[1] = av.y * sc;

        #pragma unroll
        for (int j = 0; j < 4; ++j) {
            // B operand: one contiguous ds_load_b64 = {W[kb][n], W[kb+1][n]}
            v2f b = wp[kp * 64 + j * 16 + rIdx];
            acc[j] = __builtin_amdgcn_wmma_f32_16x16x4_f32(
                false, a, false, b, (short)0, acc[j], false, false);
        }
    }

    // Epilogue: C/D layout -> VGPR i holds M = i + 8*half, N = rIdx.
    // Columns: tiles 0..2 always < 50; tile 3 valid iff rIdx < 2.
    const float b0 = blds[     rIdx];
    const float b1 = blds[16 + rIdx];
    const float b2 = blds[32 + rIdx];
    const float b3 = blds[48 + rIdx];
    float* obase = out + (size_t)r * D_DIM;
    const bool tailCol = (rIdx < 2);

    #pragma unroll
    for (int i = 0; i < 8; ++i) {
        int orow = rowBase + i + half * 8;
        if (orow < nnode) {
            float* op = obase + (size_t)orow * OUT_STRIDE;
            __builtin_nontemporal_store(acc[0][i] + b0, op +      rIdx);
            __builtin_nontemporal_store(acc[1][i] + b1, op + 16 + rIdx);
            __builtin_nontemporal_store(acc[2][i] + b2, op + 32 + rIdx);
            if (tailCol)
                __builtin_nontemporal_store(acc[3][i] + b3, op + 48 + rIdx);
        }
    }
}

extern "C" void kernel_launch(void* const* d_in, const int* in_sizes, int n_in,
                              void* d_out, int out_size, void* d_ws, size_t ws_size,
                              hipStream_t stream) {
    const int*   edges_src   = (const int*)  d_in[0];
    const int*   edges_dst   = (const int*)  d_in[1];
    const float* feat_u_fwd  = (const float*)d_in[2];
    const float* feat_i_fwd  = (const float*)d_in[3];
    const float* feat_i_rev  = (const float*)d_in[4];
    const float* feat_u_rev  = (const float*)d_in[5];
    const float* w_self_fwd  = (const float*)d_in[6];
    const float* w_neigh_fwd = (const float*)d_in[7];
    const float* b_fwd       = (const float*)d_in[8];
    const float* w_self_rev  = (const float*)d_in[9];
    const float* w_neigh_rev = (const float*)d_in[10];
    const float* b_rev       = (const float*)d_in[11];
    float* out = (float*)d_out;
    float* ws  = (float*)d_ws;

    const size_t ws_floats = (size_t)R_REL * M_N * F_DIM + (size_t)R_REL * M_N
                           + (size_t)R_REL * U_N * F_DIM + (size_t)R_REL * U_N;
    hipMemsetAsync(d_ws, 0, ws_floats * sizeof(float), stream);

    // Phase 1: scatter (both directions fused in grid.y)
    dim3 eb(256), eg((E_N + 255) / 256, 2 * R_REL);
    edge_scatter_kernel<<<eg, eb, 0, stream>>>(edges_src, edges_dst,
                                               feat_u_fwd, feat_i_rev, ws);

    float* s_fwd   = ws;
    float* cnt_fwd = s_fwd + (size_t)R_REL * M_N * F_DIM;
    float* s_rev   = cnt_fwd + (size_t)R_REL * M_N;
    float* cnt_rev = s_rev + (size_t)R_REL * U_N * F_DIM;

    // Phase 2: users (item->user rev conv), output rows [0, U)
    dim3 gb(128);
    dim3 gu((U_N + 63) / 64, R_REL);
    sage_wmma_kernel<<<gu, gb, 0, stream>>>(feat_u_rev, s_rev, cnt_rev,
                                            w_self_rev, w_neigh_rev, b_rev,
                                            out, U_N);
    // Phase 2: items (user->item fwd conv), output rows after ufeat
    dim3 gi((M_N + 63) / 64, R_REL);
    sage_wmma_kernel<<<gi, gb, 0, stream>>>(feat_i_fwd, s_fwd, cnt_fwd,
                                            w_self_fwd, w_neigh_fwd, b_fwd,
                                            out + (size_t)U_N * OUT_STRIDE, M_N);
}